// FFTBias2d_36438502540201
// MI455X (gfx1250) — compile-verified
//
#include <hip/hip_runtime.h>
#include <hip/hip_bf16.h>
#include <math.h>

typedef __attribute__((ext_vector_type(2))) float v2f;
typedef __attribute__((ext_vector_type(4))) float v4f;
typedef __attribute__((ext_vector_type(8))) float v8f;

#define Bdim 16
#define Hdim 16
#define Edim 64
#define Sdim 48
#define NPAD 95     // 2*S-1 (rfft length)
#define KPAD 96     // GEMM K padded to multiple of 4
#define BH   256    // B*H
#define SS   2304   // S*S
#define TLEN 2306   // S*S + 2
#define PI_F 3.14159265358979323846f

// ---------------------------------------------------------------------------
// Kernel 1: Zmat[m*BH + bh] = exp(z_circ[h][m] - offset[b][h]);  row m=95 is 0
// z_circ = [w47, w47..w1, w0..w46]  (length 95)
// ---------------------------------------------------------------------------
__global__ void k_build_z(const float* __restrict__ w,       // (1,H,48)
                          const float* __restrict__ offset,  // (B,H)
                          float* __restrict__ Zmat) {        // (96 x 256)
  int m  = blockIdx.x;       // 0..95
  int bh = threadIdx.x;      // 0..255
  float val = 0.f;
  if (m < NPAD) {
    int h = bh & (Hdim - 1);
    int wi;
    if (m == 0)            wi = Sdim - 1;   // w[-1]
    else if (m <= Sdim-1)  wi = Sdim - m;   // reversed w[1:]
    else                   wi = m - Sdim;   // w[:-1]
    val = expf(w[h * Sdim + wi] - offset[bh]);
  }
  Zmat[m * BH + bh] = val;
}

// ---------------------------------------------------------------------------
// Kernel 2: fixed irfft94(rfft95(.)*.) kernel, K2[kj*96 + m], col m=95 is 0
// K[k,j,m] = (1/94)*(1 + (-1)^k cos(2pi*47*r/95)
//                      + 2*sum_{f=1..46} cos(2pi f (k/94 - r/95))),  r = m+47+j
// ---------------------------------------------------------------------------
__global__ void k_build_K(float* __restrict__ K2) {
  int gid = blockIdx.x * blockDim.x + threadIdx.x;  // 0 .. 2304*96-1
  int kj = gid / KPAD;
  int m  = gid - kj * KPAD;
  float val = 0.f;
  if (m < NPAD) {
    int k = kj / Sdim;
    int j = kj - k * Sdim;
    int r = m + (Sdim - 1) + j;
    float x = (float)k * (1.0f / 94.0f) - (float)r * (1.0f / 95.0f);
    float c47 = cosf(2.0f * PI_F * 47.0f * (float)r * (1.0f / 95.0f));
    float acc = 1.0f + ((k & 1) ? -c47 : c47);
    for (int f = 1; f <= 46; ++f)
      acc += 2.0f * cosf(2.0f * PI_F * (float)f * x);
    val = acc * (1.0f / 94.0f);
  }
  K2[kj * KPAD + m] = val;
}

// ---------------------------------------------------------------------------
// Kernel 3: T_all[bh][kj] = sum_m K2[kj][m] * Zmat[m][bh]
// GEMM M=2304(kj) N=256(bh) K=96(m) via V_WMMA_F32_16X16X4_F32, 1 wave/tile.
// ---------------------------------------------------------------------------
__global__ void k_gemm_T(const float* __restrict__ K2,    // (2304 x 96)
                         const float* __restrict__ Zmat,  // (96 x 256)
                         float* __restrict__ Tall) {      // (256 x 2304)
  int m0   = blockIdx.x * 16;           // over kj
  int n0   = blockIdx.y * 16;           // over bh
  int lane = threadIdx.x;               // 0..31 (wave32)
  int Ml   = lane & 15;                 // A row / B,D col
  int koff = (lane >> 4) << 1;          // K sub-offset: lanes 16-31 hold K=2,3
  int mhi  = (lane >> 4) << 3;          // D: lanes 16-31 hold M=8..15
  v8f c = {};
  for (int kb = 0; kb < KPAD; kb += 4) {
    v2f a, b;
    const float* ap = K2 + (m0 + Ml) * KPAD + kb + koff;
    a.x = ap[0];
    a.y = ap[1];
    b.x = Zmat[(kb + koff) * BH + n0 + Ml];
    b.y = Zmat[(kb + koff + 1) * BH + n0 + Ml];
    c = __builtin_amdgcn_wmma_f32_16x16x4_f32(false, a, false, b,
                                              (short)0, c, false, false);
  }
  for (int r = 0; r < 8; ++r)
    Tall[(size_t)(n0 + Ml) * SS + (m0 + r + mhi)] = c[r];
}

// ---------------------------------------------------------------------------
// Kernel 4: one block per (b,h). Single b128-vectorized pass over v ->
// vsum in registers (exclusive j-ownership), usum via 1 LDS atomic per i,
// Rv = T@vsum, Ru = T@usum via WMMA, stream out1/out2 with b128 stores.
// ---------------------------------------------------------------------------
__global__ __launch_bounds__(256) void k_main(
    const float* __restrict__ v,     // (B, 2306, H, E)
    const float* __restrict__ Tall,  // (256 x 2304)
    float* __restrict__ out1,        // (B, 2306, H, E)
    float* __restrict__ out2) {      // (B, 2306, H)
  __shared__ float Ts[SS];                 //  9.0 KB  T[k][j]
  __shared__ float vsumS[Sdim * Edim];     // 12.0 KB  [j][e]
  __shared__ float usumS[Sdim * Edim];     // 12.0 KB  [i][e]
  __shared__ float RvS[Sdim * Edim];       // 12.0 KB  [k][e]
  __shared__ float RuS[Sdim * Edim];       // 12.0 KB  [k][e]
  __shared__ float zp[Sdim];

  int bh  = blockIdx.x;
  int b   = bh >> 4;
  int h   = bh & 15;
  int tid = threadIdx.x;

  // stage T (coalesced: T_all is [bh][kj]); zero the shared accumulator
  for (int i = tid; i < SS; i += 256) Ts[i] = Tall[(size_t)bh * SS + i];
  for (int i = tid; i < Sdim * Edim; i += 256) usumS[i] = 0.f;
  __syncthreads();

  // -------- single pass over v, 16 t-rows x 16 e-quads per iteration ------
  // thread owns e-quad e4..e4+3 and t-residue tg (mod 16)
  int tg = tid >> 4;           // 0..15
  int e4 = (tid & 15) << 2;    // 0,4,..,60
  const float* vb = v + (((size_t)b * TLEN + 1) * Hdim + h) * Edim + e4;

  v4f vacc[3];                 // exclusive owner of j = tg, tg+16, tg+32
  vacc[0] = v4f{}; vacc[1] = v4f{}; vacc[2] = v4f{};
  for (int i = 0; i < Sdim; ++i) {
    v4f u = {};
#pragma unroll
    for (int jj = 0; jj < 3; ++jj) {
      int idx = i * Sdim + tg + 16 * jj;            // j = tg + 16*jj
      v4f val = *(const v4f*)(vb + (size_t)idx * (Hdim * Edim));
      vacc[jj] += val;
      u += val;
    }
    atomicAdd(&usumS[i * Edim + e4 + 0], u[0]);     // ds_add_f32
    atomicAdd(&usumS[i * Edim + e4 + 1], u[1]);
    atomicAdd(&usumS[i * Edim + e4 + 2], u[2]);
    atomicAdd(&usumS[i * Edim + e4 + 3], u[3]);
  }
#pragma unroll
  for (int jj = 0; jj < 3; ++jj)
    *(v4f*)&vsumS[(tg + 16 * jj) * Edim + e4] = vacc[jj];
  __syncthreads();

  // zp[k] = 48 * sum_j T[k][j]   (for out2)
  if (tid < Sdim) {
    float acc = 0.f;
    for (int j = 0; j < Sdim; ++j) acc += Ts[tid * Sdim + j];
    zp[tid] = acc * (float)Sdim;
  }

  // -------- Rv = T @ vsum, Ru = T @ usum : 24 16x16 tiles over 8 waves -----
  int wave = tid >> 5;
  int lane = tid & 31;
  int Ml   = lane & 15;
  int koff = (lane >> 4) << 1;
  int mhi  = (lane >> 4) << 3;
  for (int tile = wave; tile < 24; tile += 8) {     // uniform per wave
    int g  = (tile >= 12) ? 1 : 0;
    int r2 = tile - g * 12;
    int m0 = (r2 % 3) * 16;                         // k-tile (48 rows)
    int n0 = (r2 / 3) * 16;                         // e-tile (64 cols)
    const float* Bs = g ? usumS : vsumS;
    float*       Ds = g ? RuS   : RvS;
    v8f c = {};
    for (int kb = 0; kb < Sdim; kb += 4) {          // K = 48 -> 12 steps
      v2f a, bb;
      a.x  = Ts[(m0 + Ml) * Sdim + kb + koff];
      a.y  = Ts[(m0 + Ml) * Sdim + kb + koff + 1];
      bb.x = Bs[(kb + koff)     * Edim + n0 + Ml];
      bb.y = Bs[(kb + koff + 1) * Edim + n0 + Ml];
      c = __builtin_amdgcn_wmma_f32_16x16x4_f32(false, a, false, bb,
                                                (short)0, c, false, false);
    }
    for (int r = 0; r < 8; ++r)
      Ds[(m0 + r + mhi) * Edim + n0 + Ml] = c[r];
  }
  __syncthreads();

  // -------- out1: interior t = 1 + 48*i + j (b128 stores); borders zero ----
  float* o1 = out1 + (((size_t)b * TLEN + 1) * Hdim + h) * Edim + e4;
  for (int idx = tg; idx < SS; idx += 16) {
    int i = idx / Sdim;
    int j = idx - i * Sdim;
    v4f rv = *(const v4f*)&RvS[j * Edim + e4];
    v4f ru = *(const v4f*)&RuS[i * Edim + e4];
    *(v4f*)(o1 + (size_t)idx * (Hdim * Edim)) = rv + ru;
  }
  if (tid < 16) {
    v4f zz = {};
    *(v4f*)(out1 + (((size_t)b * TLEN + 0)        * Hdim + h) * Edim + (tid << 2)) = zz;
    *(v4f*)(out1 + (((size_t)b * TLEN + (TLEN-1)) * Hdim + h) * Edim + (tid << 2)) = zz;
  }

  // -------- out2[b][t][h] = zp[j] + zp[i] (borders zero) --------
  for (int t = tid; t < TLEN; t += 256) {
    float val = 0.f;
    if (t >= 1 && t <= SS) {
      int idx = t - 1;
      int i = idx / Sdim;
      int j = idx - i * Sdim;
      val = zp[j] + zp[i];
    }
    out2[((size_t)b * TLEN + t) * Hdim + h] = val;
  }
}

// ---------------------------------------------------------------------------
extern "C" void kernel_launch(void* const* d_in, const int* in_sizes, int n_in,
                              void* d_out, int out_size, void* d_ws, size_t ws_size,
                              hipStream_t stream) {
  const float* v      = (const float*)d_in[0];   // (16,2306,16,64)
  const float* offset = (const float*)d_in[1];   // (16,16)
  const float* w      = (const float*)d_in[2];   // (1,16,48)

  float* out1 = (float*)d_out;                                     // 37,781,504 f32
  float* out2 = out1 + (size_t)Bdim * TLEN * Hdim * Edim;          //    590,336 f32

  float* ws   = (float*)d_ws;                    // ~3.2 MB used
  float* K2   = ws;                              // 2304*96
  float* Zmat = K2 + SS * KPAD;                  // 96*256
  float* Tall = Zmat + KPAD * BH;                // 2304*256

  k_build_z<<<KPAD, BH, 0, stream>>>(w, offset, Zmat);
  k_build_K<<<(SS * KPAD) / 256, 256, 0, stream>>>(K2);
  k_gemm_T<<<dim3(SS / 16, BH / 16), 32, 0, stream>>>(K2, Zmat, Tall);
  k_main<<<BH, 256, 0, stream>>>(v, Tall, out1, out2);

  (void)in_sizes; (void)n_in; (void)out_size; (void)ws_size;
}